// DAS_PW_69260642615651
// MI455X (gfx1250) — compile-verified
//
#include <hip/hip_runtime.h>
#include <math.h>

// ---------------------------------------------------------------------------
// Plane-wave DAS beamformer for MI455X (gfx1250, wave32).
// 16x16-pixel tile per 256-thread block (8 wave32s). For each (angle,element)
// the tile only touches a <~100-sample span of the 2048-sample I/Q rows, so a
// 128-sample window is staged into LDS with gfx1250 async global->LDS copies
// (ASYNCcnt), double-buffered; interpolation gathers run from LDS.
// Phase: FDEMOD/FS == 1/4 exactly -> theta = (pi/2)*i0 + ((pi/2)*w - phi):
// quadrant select on i0&3 + small-argument sincos.
// ---------------------------------------------------------------------------

#define NANG  75
#define NELE  128
#define NSAMP 2048
#define NXg   256
#define NZg   256
#define NPIX  (NXg * NZg)
#define WIN   128            // staged window length (samples); span bound ~100

static constexpr float FS_F    = 20832000.0f;
static constexpr float FS_C    = (float)(20832000.0 / 1540.0);   // samples/m
static constexpr float PIO2_F  = 1.5707963268f;
static constexpr float TWOPI_F = 6.2831853072f;
static constexpr float I2PI_F  = 0.1591549431f;

// LDS layout in floats. Dynamic LDS starts at byte 0, so absolute LDS byte
// addresses can be used as the async-copy VDST operand.
//   [0    .. 511 ]  2 parity buffers x (128 I + 128 Q) window floats
//   [512  .. 811 ]  per-angle float4 {sinA*FS/C, cosA*FS/C, t0*FS, min_tx}
//   [812  .. 1067]  per-element float2 {x_e, min_rx}
#define ANG4_OFF 512                 // byte 2048: 16B aligned for ds_load_b128
#define ELE2_OFF (512 + 4 * NANG)    // byte 3248: 8B aligned for ds_load_b64
#define LDS_FLOATS (ELE2_OFF + 2 * NELE)   // 1068 floats = 4272 bytes

__device__ __forceinline__ int window_start(float f) {
    int ws = (int)floorf(f) - 2;
    ws = ws < 0 ? 0 : ws;
    ws = ws > (NSAMP - WIN) ? (NSAMP - WIN) : ws;
    return ws & ~3;                  // 16-byte align the global window start
}

// Stage the (e2, a2) I/Q windows into LDS parity buffer ((e2+a2)&1).
// Wave 0 copies the I window, wave 1 the Q window: one
// global_load_async_to_lds_b128 per wave moves 32 lanes x 16B = 512B.
__device__ __forceinline__ void stage_window(int e2, int a2, int tid,
                                             const float* __restrict__ idata,
                                             const float* __restrict__ qdata,
                                             const float* __restrict__ sm) {
    if (tid >= 64) return;
    const float wsf = sm[ANG4_OFF + 4 * a2 + 3] + sm[ELE2_OFF + 2 * e2 + 1];
    const int   ws  = window_start(wsf);
    const unsigned par    = (unsigned)((e2 + a2) & 1);
    const unsigned lane16 = (unsigned)(tid & 31) * 16u;
    const unsigned voff   =
        ((unsigned)(((a2 << 7) + e2) << 11) + (unsigned)ws) * 4u + lane16;
    if (tid < 32) {
        const unsigned lds_off = par * 1024u + lane16;             // I half
        asm volatile("global_load_async_to_lds_b128 %0, %1, %2 offset:0"
                     :: "v"(lds_off), "v"(voff), "s"(idata) : "memory");
    } else {
        const unsigned lds_off = par * 1024u + 512u + lane16;      // Q half
        asm volatile("global_load_async_to_lds_b128 %0, %1, %2 offset:0"
                     :: "v"(lds_off), "v"(voff), "s"(qdata) : "memory");
    }
}

extern "C" __global__ __launch_bounds__(256)
void das_pw_kernel(const float* __restrict__ idata,
                   const float* __restrict__ qdata,
                   const float* __restrict__ angles,
                   const float* __restrict__ ele_pos,
                   const float* __restrict__ time_zero,
                   const float* __restrict__ grid,
                   float* __restrict__ out) {
    extern __shared__ float sm[];
    const int tid = threadIdx.x;

    // Pixel owned by this thread: 16(x) x 16(z) tile, z fastest for coalescing.
    const int lx = tid >> 4;
    const int lz = tid & 15;
    const int ix = (blockIdx.x << 4) + lx;
    const int iz = (blockIdx.y << 4) + lz;
    const int p  = ix * NZg + iz;
    const float px = grid[3 * p + 0];
    const float pz = grid[3 * p + 2];

    // Tile bounding rectangle (grids are monotonically increasing).
    const int ixb = blockIdx.x << 4, izb = blockIdx.y << 4;
    const float x0 = grid[3 * (ixb * NZg + izb) + 0];
    const float x1 = grid[3 * ((ixb + 15) * NZg + izb) + 0];
    const float z0 = grid[3 * (ixb * NZg + izb) + 2];

    // Per-tile constants; closed-form min over the tile rectangle.
    if (tid < NELE) {
        const float xe = ele_pos[3 * tid + 0];
        const float xc = fminf(fmaxf(xe, x0), x1);
        const float dx = xc - xe;
        float2 e2v;
        e2v.x = xe;
        e2v.y = sqrtf(fmaf(dx, dx, z0 * z0)) * FS_C;
        *(float2*)(sm + ELE2_OFF + 2 * tid) = e2v;
    } else if (tid < NELE + NANG) {
        const int   a   = tid - NELE;
        const float ang = angles[a];
        float4 a4;
        a4.x = sinf(ang) * FS_C;                       // SA
        a4.y = cosf(ang) * FS_C;                       // CA
        a4.z = time_zero[a] * FS_F;                    // T0 (samples)
        a4.w = ((a4.x >= 0.0f) ? x0 : x1) * a4.x + z0 * a4.y + a4.z;  // min_tx
        *(float4*)(sm + ANG4_OFF + 4 * a) = a4;
    }
    __syncthreads();

    // Demod phase: theta = (pi/2)*delay - phi; reduce phi mod 2*pi once.
    float phi = pz * (float)(4.0 * 3.14159265359 * 5208000.0 / 1540.0);
    phi = phi - TWOPI_F * floorf(phi * I2PI_F);
    const float mphi = -phi;

    float accI = 0.0f, accQ = 0.0f;

    // Prime the async pipeline with (e=0, a=0).
    stage_window(0, 0, tid, idata, qdata, sm);

    for (int e = 0; e < NELE; ++e) {
        const float2 E = *(const float2*)(sm + ELE2_OFF + 2 * e);  // {xe, mrx}
        const float dxe   = px - E.x;
        const float rx    = sqrtf(fmaf(dxe, dxe, pz * pz)) * FS_C;
        const float mrx_e = E.y;
        for (int a = 0; a < NANG; ++a) {
            // Wait for window t, make it visible, then kick off t+1 (which
            // overlaps with the compute below).
            asm volatile("s_wait_asynccnt 0" ::: "memory");
            __syncthreads();
            int a_n = a + 1, e_n = e;
            if (a_n == NANG) { a_n = 0; ++e_n; }
            if (e_n < NELE) stage_window(e_n, a_n, tid, idata, qdata, sm);

            // One ds_load_b128 broadcast: {SA, CA, T0, min_tx}.
            const float4 A = *(const float4*)(sm + ANG4_OFF + 4 * a);
            const int   ws = window_start(A.w + mrx_e);
            const float tx = fmaf(px, A.x, fmaf(pz, A.y, A.z));
            const float d  = tx + rx;
            const float fi = floorf(d);
            const float w  = d - fi;
            const int   i0 = (int)fi;
            int j = i0 - ws;
            j = j < 0 ? 0 : (j > (WIN - 2) ? (WIN - 2) : j);

            const float* b = sm + (((e + a) & 1) << 8);  // parity buffer
            const float v0 = b[j],        v1 = b[j + 1];
            const float u0 = b[WIN + j],  u1 = b[WIN + j + 1];
            const float it = fmaf(w, v1 - v0, v0);
            const float qt = fmaf(w, u1 - u0, u0);

            // theta = (pi/2)*i0 + alpha; quadrant-rotate small-arg sincos.
            float s, c;
            __sincosf(fmaf(w, PIO2_F, mphi), &s, &c);
            const int q = i0 & 3;
            const float cr = (q & 1) ? ((q & 2) ?  s : -s)
                                     : ((q & 2) ? -c :  c);
            const float sr = (q & 1) ? ((q & 2) ? -c :  c)
                                     : ((q & 2) ? -s :  s);

            accI = fmaf(it, cr, fmaf(qt, -sr, accI));
            accQ = fmaf(qt, cr, fmaf(it,  sr, accQ));
        }
    }

    out[p]        = accI;
    out[NPIX + p] = accQ;
}

extern "C" void kernel_launch(void* const* d_in, const int* in_sizes, int n_in,
                              void* d_out, int out_size, void* d_ws, size_t ws_size,
                              hipStream_t stream) {
    (void)in_sizes; (void)n_in; (void)out_size; (void)d_ws; (void)ws_size;
    const float* idata     = (const float*)d_in[0];
    const float* qdata     = (const float*)d_in[1];
    const float* angles    = (const float*)d_in[2];
    const float* ele_pos   = (const float*)d_in[3];
    const float* time_zero = (const float*)d_in[4];
    const float* grid      = (const float*)d_in[5];
    float* out             = (float*)d_out;

    dim3 grd(NXg / 16, NZg / 16, 1);   // 16x16 tiles of 16x16 pixels
    dim3 blk(256, 1, 1);               // 8 wave32s
    das_pw_kernel<<<grd, blk, LDS_FLOATS * sizeof(float), stream>>>(
        idata, qdata, angles, ele_pos, time_zero, grid, out);
}